// RPEConditionalTransformer_23725399343539
// MI455X (gfx1250) — compile-verified
//
#include <hip/hip_runtime.h>

// ---------------------------------------------------------------------------
// RPEConditionalTransformer for MI455X (gfx1250, wave32).
// B=1, N=M=512, D=256, H=4, DH=64, 6 blocks (self,cross)*3.
//
// Strategy:
//  * All dense GEMMs (q/k/v/o, qw=Q@Wp^T, attn@V, FF expand/squeeze) use
//    V_WMMA_F32_16X16X4_F32 (exact f32 tensor path). Each wave owns a 16x32
//    output tile (2 accumulators) so one A fragment feeds 2 WMMAs; transB is
//    a template parameter so the K-loop is branch-free.
//  * RPE is algebraically refactored: scores_rpe[h,n,m] =
//      emb[n,m,:]·qw[h,n,:] + qb[h,n], so the 256MB embedding tensors are
//    streamed exactly once per self block at ~2 flop/byte (HBM-bound,
//    ~66us floor for all 3 self blocks at 23.3 TB/s) instead of the naive
//    206-GFLOP pe projection.
//  * QK^T is fused into the same streaming kernel (k rows stay in L2).
// ---------------------------------------------------------------------------

typedef __attribute__((ext_vector_type(2))) float v2f;
typedef __attribute__((ext_vector_type(8))) float v8f;

#define SEQ   512
#define DMODEL 256
#define NHEAD 4
#define DHEAD 64
#define LN_EPS 1e-5f
#define SROW  ((size_t)SEQ * SEQ)   // per-head score plane

// ---------------------------------------------------------------------------
// f32 WMMA GEMM: C[rows x cols] = A[rows x K] * B (+bias) (opt ReLU).
// TRANSB==0: B is [K x cols] row-major (ldb). TRANSB==1: B is [cols x K] (ldb).
// One 16x32 tile per wave (2 accumulators), 4 waves/block, K-step 4.
// Fragment layouts per CDNA5 ISA 7.12.2:
//  A 16x4: lanes 0-15 M=lane, VGPR{0,1}=K{0,1}; lanes 16-31 K{2,3}.
//  B 4x16: VGPR v, lanes 0-15 -> K=v, N=lane; lanes 16-31 -> K=v+2.
//  C 16x16: VGPR v, lanes 0-15 -> M=v; lanes 16-31 -> M=v+8; N=lane&15.
// ---------------------------------------------------------------------------
template <int TRANSB>
__global__ void gemm_wmma_f32(const float* __restrict__ A, int lda,
                              const float* __restrict__ B, int ldb,
                              const float* __restrict__ bias,
                              float* __restrict__ C, int ldc,
                              int rows, int cols, int K, int relu)
{
    const int lane   = threadIdx.x;          // 0..31
    const int tilesN = cols >> 5;            // 32-wide N tiles
    const int tilesM = rows >> 4;
    const int tile   = blockIdx.x * blockDim.y + threadIdx.y;
    if (tile >= tilesN * tilesM) return;     // wave-uniform exit (EXEC all-1 for WMMA)

    const int tn   = (tile % tilesN) << 5;
    const int tm   = (tile / tilesN) << 4;
    const int half = lane >> 4;              // 0 | 1
    const int l    = lane & 15;

    v8f acc0 = {0.f, 0.f, 0.f, 0.f, 0.f, 0.f, 0.f, 0.f};
    v8f acc1 = acc0;

    const float* Arow = A + (size_t)(tm + l) * lda + 2 * half;

#pragma unroll 2
    for (int k = 0; k < K; k += 4) {
        v2f a, b0, b1;
        a.x = Arow[k];
        a.y = Arow[k + 1];
        if (TRANSB) {
            // B logical [kk][n] = B[n*ldb + kk]
            const float* Bc0 = B + (size_t)(tn + l) * ldb + k + 2 * half;
            const float* Bc1 = Bc0 + (size_t)16 * ldb;
            b0.x = Bc0[0]; b0.y = Bc0[1];
            b1.x = Bc1[0]; b1.y = Bc1[1];
        } else {
            const float* Br0 = B + (size_t)(k + 2 * half) * ldb + tn + l;
            const float* Br1 = Br0 + ldb;
            b0.x = Br0[0];  b0.y = Br1[0];
            b1.x = Br0[16]; b1.y = Br1[16];
        }
        acc0 = __builtin_amdgcn_wmma_f32_16x16x4_f32(false, a, false, b0,
                                                     (short)0, acc0, false, false);
        acc1 = __builtin_amdgcn_wmma_f32_16x16x4_f32(false, a, false, b1,
                                                     (short)0, acc1, false, false);
    }

#pragma unroll
    for (int v = 0; v < 8; ++v) {
        const int row = tm + v + 8 * half;
        float* Crow = C + (size_t)row * ldc + tn + l;
        float v0 = acc0[v];
        float v1 = acc1[v];
        if (bias) { v0 += bias[tn + l]; v1 += bias[tn + l + 16]; }
        if (relu) { v0 = fmaxf(v0, 0.f); v1 = fmaxf(v1, 0.f); }
        Crow[0]  = v0;
        Crow[16] = v1;
    }
}

// ---------------------------------------------------------------------------
// qb[h][n] = sum_c Q[n, 64h+c] * bp[64h+c]
// ---------------------------------------------------------------------------
__global__ void qb_kernel(const float* __restrict__ Q,
                          const float* __restrict__ bp,
                          float* __restrict__ qb)
{
    const int i = blockIdx.x * blockDim.x + threadIdx.x;   // h*512 + n
    if (i >= NHEAD * SEQ) return;
    const int h = i >> 9, n = i & (SEQ - 1);
    const float* qr = Q + (size_t)n * DMODEL + h * DHEAD;
    const float* b  = bp + h * DHEAD;
    float s = 0.f;
#pragma unroll 8
    for (int c = 0; c < DHEAD; ++c) s += qr[c] * b[c];
    qb[i] = s;
}

// ---------------------------------------------------------------------------
// Fused scores kernel:  scores[h][n][m] = Q[n,64h:]·K[m,64h:]
//                                        (+ emb[n,m,:]·qw[h,n,:] + qb[h,n])
// Block per n (512 blocks), 256 threads = 8 waves; wave w handles rows m=w,w+8,...
// Lane L covers d = 4L..4L+3 (chunk0) and 128+4L.. (chunk1): coalesced float4.
// QK partials live per-head-group (lanes 0-15 / 16-31); RPE reduced wave-wide.
// This kernel streams the 256MB embedding tensor -> HBM bandwidth bound.
// ---------------------------------------------------------------------------
__global__ void scores_kernel(const float* __restrict__ Q,
                              const float* __restrict__ Kmat,
                              const float* __restrict__ qw,   // [n][h][d] : n*1024 + h*256 + d
                              const float* __restrict__ qb,   // [h][n]
                              const float* __restrict__ emb,  // [n][m][256]
                              float* __restrict__ scores,     // [h][n][m]
                              int useRpe)
{
    __shared__ float q_lds[DMODEL];
    __shared__ float qw_lds[NHEAD * DMODEL];
    __shared__ float qb_lds[NHEAD];

    const int n = blockIdx.x;
    const int t = threadIdx.x;

    q_lds[t] = Q[(size_t)n * DMODEL + t];
    if (useRpe) {
#pragma unroll
        for (int j = 0; j < NHEAD; ++j)
            qw_lds[j * DMODEL + t] = qw[(size_t)n * (NHEAD * DMODEL) + j * DMODEL + t];
        if (t < NHEAD) qb_lds[t] = qb[t * SEQ + n];
    }
    __syncthreads();

    const int wave = t >> 5;
    const int lane = t & 31;
    const int d0 = 4 * lane;
    const int d1 = 128 + 4 * lane;

    for (int m = wave; m < SEQ; m += 8) {
        const float* krow = Kmat + (size_t)m * DMODEL;
        const float4 k0 = *(const float4*)(krow + d0);
        const float4 k1 = *(const float4*)(krow + d1);

        float accA = k0.x * q_lds[d0]     + k0.y * q_lds[d0 + 1]
                   + k0.z * q_lds[d0 + 2] + k0.w * q_lds[d0 + 3];
        float accB = k1.x * q_lds[d1]     + k1.y * q_lds[d1 + 1]
                   + k1.z * q_lds[d1 + 2] + k1.w * q_lds[d1 + 3];

        float r0 = 0.f, r1 = 0.f, r2 = 0.f, r3 = 0.f;
        if (useRpe) {
            const float* erow = emb + ((size_t)n * SEQ + m) * DMODEL;
            const float4 e0 = *(const float4*)(erow + d0);
            const float4 e1 = *(const float4*)(erow + d1);
            if (m + 8 < SEQ)   // gfx1250 global_prefetch_b8 on next streamed row
                __builtin_prefetch(emb + ((size_t)n * SEQ + m + 8) * DMODEL + d0, 0, 0);
#pragma unroll
            for (int h = 0; h < NHEAD; ++h) {
                const float* w = qw_lds + h * DMODEL;
                const float rh = e0.x * w[d0]     + e0.y * w[d0 + 1]
                               + e0.z * w[d0 + 2] + e0.w * w[d0 + 3]
                               + e1.x * w[d1]     + e1.y * w[d1 + 1]
                               + e1.z * w[d1 + 2] + e1.w * w[d1 + 3];
                if      (h == 0) r0 = rh;
                else if (h == 1) r1 = rh;
                else if (h == 2) r2 = rh;
                else             r3 = rh;
            }
        }

        // QK: reduce inside each 16-lane group (chunk0 -> heads 0/1, chunk1 -> 2/3)
#pragma unroll
        for (int off = 8; off >= 1; off >>= 1) {
            accA += __shfl_xor(accA, off, 32);
            accB += __shfl_xor(accB, off, 32);
        }
        if (useRpe) {
#pragma unroll
            for (int off = 16; off >= 1; off >>= 1) {   // RPE: full-wave reduce
                r0 += __shfl_xor(r0, off, 32);
                r1 += __shfl_xor(r1, off, 32);
                r2 += __shfl_xor(r2, off, 32);
                r3 += __shfl_xor(r3, off, 32);
            }
        }

        const size_t o = (size_t)n * SEQ + m;
        if (lane == 0)  scores[0 * SROW + o] = accA + (useRpe ? r0 + qb_lds[0] : 0.f);
        if (lane == 16) scores[1 * SROW + o] = accA + (useRpe ? r1 + qb_lds[1] : 0.f);
        if (lane == 1)  scores[2 * SROW + o] = accB + (useRpe ? r2 + qb_lds[2] : 0.f);
        if (lane == 17) scores[3 * SROW + o] = accB + (useRpe ? r3 + qb_lds[3] : 0.f);
    }
}

// ---------------------------------------------------------------------------
// In-place softmax over 512-wide rows with 1/sqrt(DH)=1/8 scaling.
// Block per (h,n) row, 256 threads (2 elems each).
// ---------------------------------------------------------------------------
__global__ void softmax_kernel(float* __restrict__ s)
{
    __shared__ float red[256];
    float* p = s + (size_t)blockIdx.x * SEQ;
    const int t = threadIdx.x;

    const float a = p[t], b = p[t + 256];
    red[t] = fmaxf(a, b);
    __syncthreads();
    for (int st = 128; st > 0; st >>= 1) { if (t < st) red[t] = fmaxf(red[t], red[t + st]); __syncthreads(); }
    const float mx = red[0];
    __syncthreads();

    const float scale = 0.125f;   // 1/sqrt(64)
    const float ea = __expf((a - mx) * scale);
    const float eb = __expf((b - mx) * scale);
    red[t] = ea + eb;
    __syncthreads();
    for (int st = 128; st > 0; st >>= 1) { if (t < st) red[t] += red[t + st]; __syncthreads(); }
    const float inv = 1.f / red[0];

    p[t]       = ea * inv;
    p[t + 256] = eb * inv;
}

// ---------------------------------------------------------------------------
// out[row,:] = LayerNorm(x[row,:] + resid[row,:]) * g + b   (row length 256)
// ---------------------------------------------------------------------------
__global__ void add_ln_kernel(const float* __restrict__ x,
                              const float* __restrict__ resid,
                              const float* __restrict__ g,
                              const float* __restrict__ b,
                              float* __restrict__ out)
{
    __shared__ float red[DMODEL];
    const int row = blockIdx.x, t = threadIdx.x;
    const size_t i = (size_t)row * DMODEL + t;

    const float v = x[i] + resid[i];
    red[t] = v;
    __syncthreads();
    for (int st = 128; st > 0; st >>= 1) { if (t < st) red[t] += red[t + st]; __syncthreads(); }
    const float mu = red[0] * (1.f / DMODEL);
    __syncthreads();

    const float dv = v - mu;
    red[t] = dv * dv;
    __syncthreads();
    for (int st = 128; st > 0; st >>= 1) { if (t < st) red[t] += red[t + st]; __syncthreads(); }
    const float var = red[0] * (1.f / DMODEL);

    out[i] = dv * rsqrtf(var + LN_EPS) * g[t] + b[t];
}

// ---------------------------------------------------------------------------
// Host side
// ---------------------------------------------------------------------------
struct Lin { const float* b; const float* w; };
struct Blk {
    Lin expand, k, o, p, q, squeeze, v;
    const float *ln1b, *ln1g, *ln2b, *ln2g;
    bool has_p;
};

extern "C" void kernel_launch(void* const* d_in, const int* in_sizes, int n_in,
                              void* d_out, int out_size, void* d_ws, size_t ws_size,
                              hipStream_t stream)
{
    (void)in_sizes; (void)n_in; (void)out_size; (void)ws_size;

    const float* feats0 = (const float*)d_in[0];
    const float* feats1 = (const float*)d_in[1];
    const float* emb0   = (const float*)d_in[2];
    const float* emb1   = (const float*)d_in[3];

    // params: JAX pytree leaf order -> list order, then sorted dict keys.
    // Per block: expand{b,w}, k{b,w}, ln1{b,g}, ln2{b,g}, o{b,w}, [p{b,w}],
    //            q{b,w}, squeeze{b,w}, v{b,w}
    int idx = 4;
    Blk blk[6];
    for (int i = 0; i < 6; ++i) {
        const bool self_ = (i % 2 == 0);   // BLOCKS = self,cross,self,cross,self,cross
        Blk& P = blk[i];
        P.has_p = self_;
        P.expand.b  = (const float*)d_in[idx++]; P.expand.w  = (const float*)d_in[idx++];
        P.k.b       = (const float*)d_in[idx++]; P.k.w       = (const float*)d_in[idx++];
        P.ln1b      = (const float*)d_in[idx++]; P.ln1g      = (const float*)d_in[idx++];
        P.ln2b      = (const float*)d_in[idx++]; P.ln2g      = (const float*)d_in[idx++];
        P.o.b       = (const float*)d_in[idx++]; P.o.w       = (const float*)d_in[idx++];
        if (self_) { P.p.b = (const float*)d_in[idx++]; P.p.w = (const float*)d_in[idx++]; }
        else       { P.p.b = nullptr;                    P.p.w = nullptr; }
        P.q.b       = (const float*)d_in[idx++]; P.q.w       = (const float*)d_in[idx++];
        P.squeeze.b = (const float*)d_in[idx++]; P.squeeze.w = (const float*)d_in[idx++];
        P.v.b       = (const float*)d_in[idx++]; P.v.w       = (const float*)d_in[idx++];
    }

    // Workspace carve-up (floats). Total ~11.6 MB.
    const size_t FD = (size_t)SEQ * DMODEL;          // 131072
    float* ws = (float*)d_ws;
    float* f0 = ws;                 // feats0 state
    float* f1 = f0 + FD;            // feats1 state
    float* Qb = f1 + FD;
    float* Kb = Qb + FD;
    float* Vb = Kb + FD;
    float* QW = Vb + FD;            // [n][h][d] 4*FD
    float* QB = QW + 4 * FD;        // [h][n]    2048
    float* S  = QB + NHEAD * SEQ;   // scores/attn [h][n][m] 1048576
    float* Hb = S  + NHEAD * SROW;
    float* T  = Hb + FD;
    float* X1 = T  + FD;
    float* FF = X1 + FD;            // 512x512

    auto gemm = [&](const float* A, int lda, const float* B, int ldb, int transB,
                    const float* bias, float* C, int ldc,
                    int rows, int cols, int K, int relu) {
        const int tiles = (rows / 16) * (cols / 32);
        const dim3 grid((tiles + 3) / 4), block(32, 4);
        if (transB)
            gemm_wmma_f32<1><<<grid, block, 0, stream>>>(A, lda, B, ldb, bias, C, ldc,
                                                         rows, cols, K, relu);
        else
            gemm_wmma_f32<0><<<grid, block, 0, stream>>>(A, lda, B, ldb, bias, C, ldc,
                                                         rows, cols, K, relu);
    };

    auto run_block = [&](const Blk& P, float* xq, const float* xkv, const float* emb) {
        // Q/K/V projections (f32 WMMA)
        gemm(xq,  DMODEL, P.q.w, DMODEL, 0, P.q.b, Qb, DMODEL, SEQ, DMODEL, DMODEL, 0);
        gemm(xkv, DMODEL, P.k.w, DMODEL, 0, P.k.b, Kb, DMODEL, SEQ, DMODEL, DMODEL, 0);
        gemm(xkv, DMODEL, P.v.w, DMODEL, 0, P.v.b, Vb, DMODEL, SEQ, DMODEL, DMODEL, 0);

        const int useRpe = (emb != nullptr);
        if (useRpe) {
            // qw[h] = Q_h (512x64) @ Wp_h^T (64x256) -> stored [n][h][d]
            for (int h = 0; h < NHEAD; ++h)
                gemm(Qb + h * DHEAD, DMODEL, P.p.w + h * DHEAD, DMODEL, 1,
                     nullptr, QW + h * DMODEL, NHEAD * DMODEL, SEQ, DMODEL, DHEAD, 0);
            qb_kernel<<<(NHEAD * SEQ + 255) / 256, 256, 0, stream>>>(Qb, P.p.b, QB);
        }

        // Fused QK^T (+RPE streaming) then softmax(x/8)
        scores_kernel<<<SEQ, 256, 0, stream>>>(Qb, Kb, QW, QB, emb, S, useRpe);
        softmax_kernel<<<NHEAD * SEQ, 256, 0, stream>>>(S);

        // attn @ V per head, packed back to [n][h*64+c]
        for (int h = 0; h < NHEAD; ++h)
            gemm(S + (size_t)h * SROW, SEQ, Vb + h * DHEAD, DMODEL, 0,
                 nullptr, Hb + h * DHEAD, DMODEL, SEQ, DHEAD, SEQ, 0);

        // output projection + residual LN1
        gemm(Hb, DMODEL, P.o.w, DMODEL, 0, P.o.b, T, DMODEL, SEQ, DMODEL, DMODEL, 0);
        add_ln_kernel<<<SEQ, DMODEL, 0, stream>>>(T, xq, P.ln1g, P.ln1b, X1);

        // FF: expand(+ReLU) -> squeeze -> residual LN2 (writes xq in place)
        gemm(X1, DMODEL, P.expand.w, 2 * DMODEL, 0, P.expand.b, FF, 2 * DMODEL,
             SEQ, 2 * DMODEL, DMODEL, 1);
        gemm(FF, 2 * DMODEL, P.squeeze.w, DMODEL, 0, P.squeeze.b, T, DMODEL,
             SEQ, DMODEL, 2 * DMODEL, 0);
        add_ln_kernel<<<SEQ, DMODEL, 0, stream>>>(T, X1, P.ln2g, P.ln2b, xq);
    };

    hipMemcpyAsync(f0, feats0, FD * sizeof(float), hipMemcpyDeviceToDevice, stream);
    hipMemcpyAsync(f1, feats1, FD * sizeof(float), hipMemcpyDeviceToDevice, stream);

    for (int i = 0; i < 6; ++i) {
        if (blk[i].has_p) {                         // self
            run_block(blk[i], f0, f0, emb0);
            run_block(blk[i], f1, f1, emb1);
        } else {                                    // cross (sequential: f1 sees updated f0)
            run_block(blk[i], f0, f1, nullptr);
            run_block(blk[i], f1, f0, nullptr);
        }
    }

    float* out = (float*)d_out;
    hipMemcpyAsync(out,      f0, FD * sizeof(float), hipMemcpyDeviceToDevice, stream);
    hipMemcpyAsync(out + FD, f1, FD * sizeof(float), hipMemcpyDeviceToDevice, stream);
}